// InceptionGATv2Block_75161927680554
// MI455X (gfx1250) — compile-verified
//
#include <hip/hip_runtime.h>
#include <hip/hip_bf16.h>

// Problem constants (match reference)
#define NN 50000
#define EE 800000
#define ETOT (EE + NN)   // 850000, self loops appended
#define INDIM 128
#define OUTDIM 128
#define HEADS 4
#define CH 32            // per-head channels
#define PP 3
#define NEGS 0.2f
#define LNEPS 1e-5f

typedef float v2f __attribute__((ext_vector_type(2)));
typedef float v8f __attribute__((ext_vector_type(8)));

// ---- ordered-uint encoding for float atomicMax (handles negatives) ----
__device__ __forceinline__ unsigned f2ord(float f) {
    unsigned u = __float_as_uint(f);
    return (u & 0x80000000u) ? ~u : (u | 0x80000000u);
}
__device__ __forceinline__ float ord2f(unsigned u) {
    u = (u & 0x80000000u) ? (u & 0x7fffffffu) : ~u;
    return __uint_as_float(u);
}

// ---------------------------------------------------------------------
// Kernel 0: init d_out accumulator, smax (ordered-min), ssum
// ---------------------------------------------------------------------
__global__ void k_init(float* __restrict__ out, unsigned* __restrict__ smax,
                       float* __restrict__ ssum) {
    size_t i = (size_t)blockIdx.x * blockDim.x + threadIdx.x;
    const size_t nOut = (size_t)NN * (PP * OUTDIM);
    if (i < nOut) out[i] = 0.0f;
    const size_t nSeg = (size_t)PP * NN * HEADS;
    if (i < nSeg) { smax[i] = 0u; ssum[i] = 0.0f; }
}

// ---------------------------------------------------------------------
// Kernel 1: xl[p] = x @ Wl[p] + bl[p]; xr[p] = x @ Wr[p] + br[p]
// One wave per 16x16 output tile, fp32 WMMA 16x16x4, K=128 in 32 steps.
// A frag: lane<16 -> rows M=lane, K={k,k+1}; lane>=16 -> K={k+2,k+3}
// B frag: lane<16 -> col N=lane, K={k,k+1}; lane>=16 -> K={k+2,k+3}
// C/D:    lane<16 -> M=r (vgpr r), lane>=16 -> M=8+r; N = lane&15
// ---------------------------------------------------------------------
__global__ void k_gemm_wmma(const float* __restrict__ x,
                            const float* __restrict__ Wl, const float* __restrict__ bl,
                            const float* __restrict__ Wr, const float* __restrict__ br,
                            float* __restrict__ xl, float* __restrict__ xr) {
    const int wavesPerBlock = blockDim.x >> 5;
    const int gw   = blockIdx.x * wavesPerBlock + (threadIdx.x >> 5);
    const int lane = threadIdx.x & 31;

    const int tilesM = NN / 16;            // 3125
    const int tilesN = OUTDIM / 16;        // 8
    const int tilesPerMat = tilesM * tilesN;

    const int mat = gw / tilesPerMat;      // 0..5 -> (path, l/r)
    if (mat >= 2 * PP) return;
    const int t  = gw % tilesPerMat;
    const int tm = t / tilesN;
    const int tn = t % tilesN;
    const int p  = mat >> 1;
    const bool isR = (mat & 1);

    const float* __restrict__ W    = (isR ? Wr : Wl) + (size_t)p * INDIM * OUTDIM;
    const float* __restrict__ bias = (isR ? br : bl) + (size_t)p * OUTDIM;
    float* __restrict__ dst        = (isR ? xr : xl) + (size_t)p * NN * OUTDIM;

    const int half = lane >> 4;            // 0 or 1
    const int l15  = lane & 15;
    const int row  = tm * 16 + l15;        // A row for this lane
    const int col  = tn * 16 + l15;        // B/D column for this lane
    const float* __restrict__ xrow = x + (size_t)row * INDIM;

    v8f c = {0.f, 0.f, 0.f, 0.f, 0.f, 0.f, 0.f, 0.f};
    #pragma unroll 4
    for (int k = 0; k < INDIM; k += 4) {
        const int kb = k + half * 2;
        v2f a, b;
        a.x = xrow[kb];
        a.y = xrow[kb + 1];
        b.x = W[(size_t)kb * OUTDIM + col];
        b.y = W[(size_t)(kb + 1) * OUTDIM + col];
        c = __builtin_amdgcn_wmma_f32_16x16x4_f32(false, a, false, b,
                                                  (short)0, c, false, false);
    }

    const float bv = bias[col];
    #pragma unroll
    for (int r = 0; r < 8; ++r) {
        const int m = tm * 16 + half * 8 + r;
        dst[(size_t)m * OUTDIM + col] = c[r] + bv;
    }
}

// ---------------------------------------------------------------------
// Kernel 2: per (path, edge) attention score + segment max
// One wave per (p, e); lane owns 4 channels; head = lane>>3.
// ---------------------------------------------------------------------
__global__ void k_score(const float* __restrict__ xl, const float* __restrict__ xr,
                        const long long* __restrict__ ei,
                        const float* __restrict__ att,
                        float* __restrict__ sc, unsigned* __restrict__ smax) {
    const int gw   = blockIdx.x * (blockDim.x >> 5) + (threadIdx.x >> 5);
    const int lane = threadIdx.x & 31;
    const long long total = (long long)PP * ETOT;
    if (gw >= total) return;
    const int p = gw / ETOT;
    const long long e = gw % ETOT;
    long long s, d;
    if (e < EE) { s = ei[e]; d = ei[EE + e]; } else { s = d = e - EE; }

    const float4 lv = *(const float4*)(xl + ((size_t)p * NN + s) * OUTDIM + lane * 4);
    const float4 rv = *(const float4*)(xr + ((size_t)p * NN + d) * OUTDIM + lane * 4);
    const float4 av = *(const float4*)(att + (size_t)p * OUTDIM + lane * 4);

    float m0 = lv.x + rv.x, m1 = lv.y + rv.y, m2 = lv.z + rv.z, m3 = lv.w + rv.w;
    float partial = (m0 > 0.f ? m0 : NEGS * m0) * av.x
                  + (m1 > 0.f ? m1 : NEGS * m1) * av.y
                  + (m2 > 0.f ? m2 : NEGS * m2) * av.z
                  + (m3 > 0.f ? m3 : NEGS * m3) * av.w;
    // reduce within 8-lane head group
    partial += __shfl_xor(partial, 1);
    partial += __shfl_xor(partial, 2);
    partial += __shfl_xor(partial, 4);

    if ((lane & 7) == 0) {
        const int h = lane >> 3;
        sc[((size_t)p * ETOT + e) * HEADS + h] = partial;
        atomicMax(&smax[((size_t)p * NN + d) * HEADS + h], f2ord(partial));
    }
}

// ---------------------------------------------------------------------
// Kernel 3: ex = exp(score - smax[dst]); ssum[dst] += ex (in-place on sc)
// One thread per (p, e, h).
// ---------------------------------------------------------------------
__global__ void k_expsum(float* __restrict__ sc, const unsigned* __restrict__ smax,
                         float* __restrict__ ssum, const long long* __restrict__ ei) {
    const long long idx = (long long)blockIdx.x * blockDim.x + threadIdx.x;
    const long long total = (long long)PP * ETOT * HEADS;
    if (idx >= total) return;
    const int h = (int)(idx & (HEADS - 1));
    const long long pe = idx >> 2;
    const long long e = pe % ETOT;
    const int p = (int)(pe / ETOT);
    const long long d = (e < EE) ? ei[EE + e] : (e - EE);

    const float mx = ord2f(smax[((size_t)p * NN + d) * HEADS + h]);
    const float ex = __expf(sc[idx] - mx);
    sc[idx] = ex;
    atomicAdd(&ssum[((size_t)p * NN + d) * HEADS + h], ex);
}

// ---------------------------------------------------------------------
// Kernel 4: out[dst] += (ex/ssum[dst]) * xl[src], scatter via atomics
// One wave per (p, e); lane owns 4 channels.
// ---------------------------------------------------------------------
__global__ void k_aggregate(const float* __restrict__ xl, const float* __restrict__ sc,
                            const float* __restrict__ ssum,
                            const long long* __restrict__ ei, float* __restrict__ out) {
    const int gw   = blockIdx.x * (blockDim.x >> 5) + (threadIdx.x >> 5);
    const int lane = threadIdx.x & 31;
    const long long total = (long long)PP * ETOT;
    if (gw >= total) return;
    const int p = gw / ETOT;
    const long long e = gw % ETOT;
    long long s, d;
    if (e < EE) { s = ei[e]; d = ei[EE + e]; } else { s = d = e - EE; }

    const int h = lane >> 3;
    const float ex  = sc[((size_t)p * ETOT + e) * HEADS + h];
    const float sm  = ssum[((size_t)p * NN + d) * HEADS + h];
    const float alpha = ex / sm;

    const float4 lv = *(const float4*)(xl + ((size_t)p * NN + s) * OUTDIM + lane * 4);
    float* o = out + (size_t)d * (PP * OUTDIM) + p * OUTDIM + lane * 4;
    atomicAdd(o + 0, alpha * lv.x);
    atomicAdd(o + 1, alpha * lv.y);
    atomicAdd(o + 2, alpha * lv.z);
    atomicAdd(o + 3, alpha * lv.w);
}

// ---------------------------------------------------------------------
// Kernel 5: h = relu(acc + out_bias); LayerNorm over 384 feats, in place.
// One block (128 threads) per node; each thread owns 3 features.
// ---------------------------------------------------------------------
__global__ void k_lnorm(float* __restrict__ out, const float* __restrict__ bias,
                        const float* __restrict__ gamma, const float* __restrict__ beta) {
    const int n = blockIdx.x;
    const int t = threadIdx.x;          // 0..127
    const int F = PP * OUTDIM;          // 384
    __shared__ float sh1[128];
    __shared__ float sh2[128];

    float v[3];
    float s = 0.f, sq = 0.f;
    #pragma unroll
    for (int k = 0; k < 3; ++k) {
        const int f = t + k * 128;
        float x = out[(size_t)n * F + f] + bias[f];
        x = x > 0.f ? x : 0.f;          // ReLU
        v[k] = x; s += x; sq += x * x;
    }
    sh1[t] = s; sh2[t] = sq;
    __syncthreads();
    for (int off = 64; off > 0; off >>= 1) {
        if (t < off) { sh1[t] += sh1[t + off]; sh2[t] += sh2[t + off]; }
        __syncthreads();
    }
    const float mean = sh1[0] * (1.0f / F);
    const float var  = sh2[0] * (1.0f / F) - mean * mean;
    const float inv  = rsqrtf(var + LNEPS);
    #pragma unroll
    for (int k = 0; k < 3; ++k) {
        const int f = t + k * 128;
        out[(size_t)n * F + f] = (v[k] - mean) * inv * gamma[f] + beta[f];
    }
}

// ---------------------------------------------------------------------
extern "C" void kernel_launch(void* const* d_in, const int* in_sizes, int n_in,
                              void* d_out, int out_size, void* d_ws, size_t ws_size,
                              hipStream_t stream) {
    const float*     x        = (const float*)d_in[0];
    const long long* ei       = (const long long*)d_in[1];   // [2, E] int64
    const float*     Wl       = (const float*)d_in[2];
    const float*     bl       = (const float*)d_in[3];
    const float*     Wr       = (const float*)d_in[4];
    const float*     br       = (const float*)d_in[5];
    const float*     att      = (const float*)d_in[6];
    const float*     out_bias = (const float*)d_in[7];
    const float*     gamma    = (const float*)d_in[8];
    const float*     beta     = (const float*)d_in[9];
    float*           out      = (float*)d_out;

    // workspace carve-up (~200 MB)
    float*    xl   = (float*)d_ws;                              // P*N*128
    float*    xr   = xl + (size_t)PP * NN * OUTDIM;             // P*N*128
    float*    sc   = xr + (size_t)PP * NN * OUTDIM;             // P*Etot*H
    unsigned* smax = (unsigned*)(sc + (size_t)PP * ETOT * HEADS); // P*N*H
    float*    ssum = (float*)(smax + (size_t)PP * NN * HEADS);    // P*N*H

    // 0) init accumulators
    {
        const size_t nOut = (size_t)NN * PP * OUTDIM;          // 19.2M (covers seg bufs too)
        const int blocks = (int)((nOut + 255) / 256);
        k_init<<<blocks, 256, 0, stream>>>(out, smax, ssum);
    }
    // 1) WMMA GEMMs: 6 matrices x 3125x8 tiles, 8 waves/block
    {
        const int totalWaves = 2 * PP * (NN / 16) * (OUTDIM / 16); // 150000
        const int blocks = (totalWaves + 7) / 8;
        k_gemm_wmma<<<blocks, 256, 0, stream>>>(x, Wl, bl, Wr, br, xl, xr);
    }
    // 2) attention scores + segment max (wave per (p,e))
    {
        const long long waves = (long long)PP * ETOT;          // 2.55M
        const int blocks = (int)((waves + 7) / 8);
        k_score<<<blocks, 256, 0, stream>>>(xl, xr, ei, att, sc, smax);
    }
    // 3) exp + segment sum (thread per (p,e,h))
    {
        const long long thr = (long long)PP * ETOT * HEADS;    // 10.2M
        const int blocks = (int)((thr + 255) / 256);
        k_expsum<<<blocks, 256, 0, stream>>>(sc, smax, ssum, ei);
    }
    // 4) alpha-weighted scatter (wave per (p,e))
    {
        const long long waves = (long long)PP * ETOT;
        const int blocks = (int)((waves + 7) / 8);
        k_aggregate<<<blocks, 256, 0, stream>>>(xl, sc, ssum, ei, out);
    }
    // 5) bias + relu + layernorm, in place on d_out
    {
        k_lnorm<<<NN, 128, 0, stream>>>(out, out_bias, gamma, beta);
    }
}